// PHOSAInteractionLoss_61830349193773
// MI455X (gfx1250) — compile-verified
//
#include <hip/hip_runtime.h>

// PHOSA interaction loss for MI455X (gfx1250, wave32).
// Pipeline (all on `stream`, dependencies via stream order, zero atomics,
// fully deterministic — partials summed in fixed order):
//   1) batch_sums / batch_sums_vec4 : chunked streaming sums -> partials in ws
//      (object_v path uses float4 (b128) loads: rows are 16B-aligned; smpl_v
//       rows are only 4B-aligned so it keeps the scalar path)
//   2) part_stats   : one WG per (b,p): gather + project, 9-way LDS tree reduction
//   3) pair_loss    : one wave32 per batch; V_WMMA_F32_16X16X4_F32 computes the
//                     8x8 part-mean dot-product table; masking + per-batch sums
//   4) finalize     : single thread, fixed-order sums -> out[2]

#define B_   256
#define NS_  10475
#define NO_  65536
#define P_   8
#define KS_  1024
#define KO_  2048
#define EPSF 1e-9f
#define ZTH  5.0f

#define SCH_ 6   // smpl chunks (6*2048 >= 10475)
#define OCH_ 8   // object chunks (8*8192 == 65536)

// ---- workspace layout (float offsets) ----
#define WS_SPART   0                         // B*SCH_*3 = 4608
#define WS_OPART   (WS_SPART + B_*SCH_*3)    // B*OCH_*3 = 6144
#define WS_SSTATS  (WS_OPART + B_*OCH_*3)    // B*P*9 = 18432
#define WS_OSTATS  (WS_SSTATS + B_*P_*9)     // B*P*9 = 18432
#define WS_INTERB  (WS_OSTATS + B_*P_*9)     // B
#define WS_PSUMB   (WS_INTERB + B_)          // B
#define WS_PCNTB   (WS_PSUMB + B_)           // B

typedef float v2f __attribute__((ext_vector_type(2)));
typedef float v8f __attribute__((ext_vector_type(8)));

// ---------- shared block-reduce helper for 3 sums ----------
__device__ __forceinline__ void reduce3_store(float sx, float sy, float sz,
                                              float* __restrict__ o3) {
  __shared__ float red[3][256];
  const int t = threadIdx.x;
  red[0][t] = sx; red[1][t] = sy; red[2][t] = sz;
  __syncthreads();
  for (int s = 128; s > 0; s >>= 1) {
    if (t < s) {
      red[0][t] += red[0][t + s];
      red[1][t] += red[1][t + s];
      red[2][t] += red[2][t + s];
    }
    __syncthreads();
  }
  if (t == 0) { o3[0] = red[0][0]; o3[1] = red[1][0]; o3[2] = red[2][0]; }
}

// ---------- kernel 1a: scalar chunked sums (rows only 4B-aligned) ----------
__global__ void batch_sums(const float* __restrict__ verts,
                           float* __restrict__ partial, // (B, nChunks, 3)
                           int N, int chunk, int nChunks) {
  const int b = blockIdx.x;
  const int cidx = blockIdx.y;
  const int start = cidx * chunk;
  int end = start + chunk; if (end > N) end = N;
  const float* vb = verts + (size_t)b * N * 3;

  float sx = 0.f, sy = 0.f, sz = 0.f;
  for (int i = start + threadIdx.x; i < end; i += blockDim.x) {
    int pf = i + (int)blockDim.x * 8;
    if (pf < end) __builtin_prefetch(vb + (size_t)pf * 3, 0, 0); // global_prefetch_b8
    const float* p = vb + (size_t)i * 3;
    sx += p[0]; sy += p[1]; sz += p[2];
  }
  reduce3_store(sx, sy, sz, partial + ((size_t)b * nChunks + cidx) * 3);
}

// ---------- kernel 1b: b128-vectorized chunked sums (16B-aligned rows) ----------
// Requires: (N*3*4) % 16 == 0, (chunk*3*4) % 16 == 0, chunk % 4 == 0, chunk | N.
// Each thread handles 4 vertices (= 3 x float4 = 48B) per iteration.
__global__ void batch_sums_vec4(const float* __restrict__ verts,
                                float* __restrict__ partial, // (B, nChunks, 3)
                                int N, int chunk, int nChunks) {
  const int b = blockIdx.x;
  const int cidx = blockIdx.y;
  const int start = cidx * chunk;           // vertex index, multiple of chunk
  const int nGroups = chunk >> 2;           // groups of 4 vertices
  const float4* v4 = (const float4*)(verts + (size_t)b * N * 3 + (size_t)start * 3);

  float sx = 0.f, sy = 0.f, sz = 0.f;
  for (int g = threadIdx.x; g < nGroups; g += blockDim.x) {
    int pf = g + (int)blockDim.x * 4;
    if (pf < nGroups) __builtin_prefetch(v4 + (size_t)pf * 3, 0, 0); // global_prefetch_b8
    const float4 a = v4[(size_t)g * 3 + 0];  // v0.xyz v1.x
    const float4 c = v4[(size_t)g * 3 + 1];  // v1.yz  v2.xy
    const float4 d = v4[(size_t)g * 3 + 2];  // v2.z   v3.xyz
    sx += (a.x + a.w) + (c.z + d.y);
    sy += (a.y + c.x) + (c.w + d.z);
    sz += (a.z + c.y) + (d.x + d.w);
  }
  reduce3_store(sx, sy, sz, partial + ((size_t)b * nChunks + cidx) * 3);
}

// ---------- kernel 2: per-(batch,part) gather stats ----------
// stats per (b,p): [umin,umax,wmin,wmax,zmin,zmax,mean_x,mean_y,mean_z]
__global__ void part_stats(const float* __restrict__ verts,
                           const float* __restrict__ camKs,   // (B,9)
                           const int*   __restrict__ partIdx, // (P,K)
                           float* __restrict__ stats,         // (B,P,9)
                           int N, int K) {
  const int b = blockIdx.x;
  const int p = blockIdx.y;
  const float fx = camKs[b * 9 + 0];
  const float fy = camKs[b * 9 + 4];
  const float cx = camKs[b * 9 + 2];
  const float cy = camKs[b * 9 + 5];
  const float* vb = verts + (size_t)b * N * 3;
  const int* pi = partIdx + (size_t)p * K;

  float umin =  INFINITY, umax = -INFINITY;
  float wmin =  INFINITY, wmax = -INFINITY;
  float zmin =  INFINITY, zmax = -INFINITY;
  float sx = 0.f, sy = 0.f, sz = 0.f;

  for (int k = threadIdx.x; k < K; k += blockDim.x) {
    const int idx = pi[k];
    const float x = vb[(size_t)idx * 3 + 0];
    const float y = vb[(size_t)idx * 3 + 1];
    const float z = vb[(size_t)idx * 3 + 2];
    const float zi = z + EPSF;
    // reference: v *= [1,-1,1]; u = 2*(fx*x/zi + cx - 0.5); w = 2*(1-(fy*(-y)/zi+cy)-0.5)
    const float u = 2.0f * (fx * (x / zi) + cx - 0.5f);
    const float w = 2.0f * ((1.0f - (fy * ((-y) / zi) + cy)) - 0.5f);
    umin = fminf(umin, u); umax = fmaxf(umax, u);
    wmin = fminf(wmin, w); wmax = fmaxf(wmax, w);
    zmin = fminf(zmin, z); zmax = fmaxf(zmax, z);
    sx += x; sy += y; sz += z;
  }

  __shared__ float red[9][256];
  const int t = threadIdx.x;
  red[0][t] = umin; red[1][t] = umax;
  red[2][t] = wmin; red[3][t] = wmax;
  red[4][t] = zmin; red[5][t] = zmax;
  red[6][t] = sx;   red[7][t] = sy;   red[8][t] = sz;
  __syncthreads();
  for (int s = 128; s > 0; s >>= 1) {
    if (t < s) {
      red[0][t] = fminf(red[0][t], red[0][t + s]);
      red[1][t] = fmaxf(red[1][t], red[1][t + s]);
      red[2][t] = fminf(red[2][t], red[2][t + s]);
      red[3][t] = fmaxf(red[3][t], red[3][t + s]);
      red[4][t] = fminf(red[4][t], red[4][t + s]);
      red[5][t] = fmaxf(red[5][t], red[5][t + s]);
      red[6][t] += red[6][t + s];
      red[7][t] += red[7][t + s];
      red[8][t] += red[8][t + s];
    }
    __syncthreads();
  }
  if (t == 0) {
    const float invK = 1.0f / (float)K;
    float* st = stats + ((size_t)b * P_ + p) * 9;
    st[0] = red[0][0]; st[1] = red[1][0];
    st[2] = red[2][0]; st[3] = red[3][0];
    st[4] = red[4][0]; st[5] = red[5][0];
    st[6] = red[6][0] * invK;
    st[7] = red[7][0] * invK;
    st[8] = red[8][0] * invK;
  }
}

// ---------- kernel 3: per-batch pair table via WMMA (one wave32 / batch) ----------
__global__ __launch_bounds__(32)
void pair_loss(const float* __restrict__ sstats,  // (B,P,9)
               const float* __restrict__ ostats,  // (B,P,9)
               const float* __restrict__ sPart,   // (B,SCH_,3)
               const float* __restrict__ oPart,   // (B,OCH_,3)
               float* __restrict__ interB,        // (B)
               float* __restrict__ psumB,         // (B)
               float* __restrict__ pcntB) {       // (B)
  const int b = blockIdx.x;
  const int t = threadIdx.x;

  __shared__ float ss[P_ * 9];
  __shared__ float os[P_ * 9];
  {
    const float* sp = sstats + (size_t)b * P_ * 9;
    const float* op = ostats + (size_t)b * P_ * 9;
    for (int i = t; i < P_ * 9; i += 32) { ss[i] = sp[i]; os[i] = op[i]; }
  }
  __syncthreads();

  // A: 16x4 f32 (rows = smpl parts, K = xyz + pad). Lanes 0-15: K=0,1; lanes 16-31: K=2,3.
  // B: 4x16 f32 (K = xyz + pad, cols = object parts). Same lane striping.
  const int M  = t & 15;
  const bool hi = t >= 16;
  v2f av, bv;
  av.x = hi ? ((M < P_) ? ss[M * 9 + 8] : 0.0f) : ((M < P_) ? ss[M * 9 + 6] : 0.0f);
  av.y = hi ? 0.0f                              : ((M < P_) ? ss[M * 9 + 7] : 0.0f);
  bv.x = hi ? ((M < P_) ? os[M * 9 + 8] : 0.0f) : ((M < P_) ? os[M * 9 + 6] : 0.0f);
  bv.y = hi ? 0.0f                              : ((M < P_) ? os[M * 9 + 7] : 0.0f);

  v8f c = {0.f, 0.f, 0.f, 0.f, 0.f, 0.f, 0.f, 0.f};
  // D[i][j] = dot(smpl_mean_i, obj_mean_j); lane j (<16) holds column j in c[0..7].
  c = __builtin_amdgcn_wmma_f32_16x16x4_f32(false, av, false, bv, (short)0, c,
                                            false, false);

  float psum = 0.0f, pcnt = 0.0f, iterm = 0.0f;

  if (t < P_) {
    const int j = t;
    // Expanded bboxes for part j (reference quirk: overlap pairs part j with part j).
    const float sul = ss[j*9+0], suh = ss[j*9+1], swl = ss[j*9+2], swh = ss[j*9+3];
    const float oul = os[j*9+0], ouh = os[j*9+1], owl = os[j*9+2], owh = os[j*9+3];
    const float p0 = 0.5f*(sul+suh) - 0.75f*(suh-sul);
    const float p1 = 0.5f*(swl+swh) - 0.75f*(swh-swl);
    const float p2 = 0.5f*(sul+suh) + 0.75f*(suh-sul);
    const float p3 = 0.5f*(swl+swh) + 0.75f*(swh-swl);
    const float o0 = 0.5f*(oul+ouh) - 0.75f*(ouh-oul);
    const float o1 = 0.5f*(owl+owh) - 0.75f*(owh-owl);
    const float o2 = 0.5f*(oul+ouh) + 0.75f*(ouh-oul);
    const float o3 = 0.5f*(owl+owh) + 0.75f*(owh-owl);
    const bool ovl = !((o0 > p2) || (p0 > o2) || (o1 > p3) || (p1 > o3));

    const float cj = os[j*9+4], dj = os[j*9+5];                     // obj z-range
    const float no = os[j*9+6]*os[j*9+6] + os[j*9+7]*os[j*9+7] + os[j*9+8]*os[j*9+8];

    #pragma unroll
    for (int i = 0; i < P_; ++i) {
      const float ai = ss[i*9+4], bi = ss[i*9+5];                   // smpl z-range
      const float gap = fminf(fabsf(cj - bi), fabsf(ai - dj));
      const float zd = ((dj >= ai) && (bi >= cj)) ? 0.0f : gap;
      if (ovl && (zd < ZTH)) {
        const float ns = ss[i*9+6]*ss[i*9+6] + ss[i*9+7]*ss[i*9+7] + ss[i*9+8]*ss[i*9+8];
        const float mse = (ns + no - 2.0f * c[i]) * (1.0f / 3.0f);
        psum += mse;
        pcnt += 1.0f;
      }
    }
  }

  if (t < 3) {
    float s = 0.0f, o = 0.0f;
    for (int k = 0; k < SCH_; ++k) s += sPart[((size_t)b * SCH_ + k) * 3 + t];
    for (int k = 0; k < OCH_; ++k) o += oPart[((size_t)b * OCH_ + k) * 3 + t];
    const float d = s * (1.0f / (float)NS_) - o * (1.0f / (float)NO_);
    iterm = d * d;
  }

  // wave32 reduction
  for (int off = 16; off > 0; off >>= 1) {
    psum  += __shfl_down(psum,  off, 32);
    pcnt  += __shfl_down(pcnt,  off, 32);
    iterm += __shfl_down(iterm, off, 32);
  }
  if (t == 0) {
    interB[b] = iterm;
    psumB[b]  = psum;
    pcntB[b]  = pcnt;
  }
}

// ---------- kernel 4: deterministic fixed-order finalize ----------
__global__ void finalize(const float* __restrict__ interB,
                         const float* __restrict__ psumB,
                         const float* __restrict__ pcntB,
                         float* __restrict__ out) {
  if (blockIdx.x == 0 && threadIdx.x == 0) {
    float inter = 0.f, ps = 0.f, pc = 0.f;
    for (int b = 0; b < B_; ++b) { inter += interB[b]; ps += psumB[b]; pc += pcntB[b]; }
    out[0] = (inter / (float)(B_ * 3)) / (float)B_;   // mean over (B,3), then / bsz
    out[1] = (pc > 0.0f) ? (ps / pc) : 0.0f;
  }
}

extern "C" void kernel_launch(void* const* d_in, const int* in_sizes, int n_in,
                              void* d_out, int out_size, void* d_ws, size_t ws_size,
                              hipStream_t stream) {
  (void)in_sizes; (void)n_in; (void)out_size; (void)ws_size;
  const float* smpl_v      = (const float*)d_in[0];
  const float* object_v    = (const float*)d_in[1];
  const float* cam_Ks      = (const float*)d_in[2];
  const int*   smpl_pidx   = (const int*)d_in[3];
  const int*   object_pidx = (const int*)d_in[4];
  float* out = (float*)d_out;
  float* ws  = (float*)d_ws;

  // 1) streaming batch sums (chunked, deterministic partials).
  //    smpl rows are 4B-aligned -> scalar; object rows are 16B-aligned -> b128 path.
  batch_sums     <<<dim3(B_, SCH_), 256, 0, stream>>>(smpl_v,   ws + WS_SPART, NS_, 2048, SCH_);
  batch_sums_vec4<<<dim3(B_, OCH_), 256, 0, stream>>>(object_v, ws + WS_OPART, NO_, 8192, OCH_);

  // 2) per-(batch,part) gather stats
  part_stats<<<dim3(B_, P_), 256, 0, stream>>>(smpl_v,   cam_Ks, smpl_pidx,
                                               ws + WS_SSTATS, NS_, KS_);
  part_stats<<<dim3(B_, P_), 256, 0, stream>>>(object_v, cam_Ks, object_pidx,
                                               ws + WS_OSTATS, NO_, KO_);

  // 3) WMMA pair table + masking, one wave32 per batch
  pair_loss<<<B_, 32, 0, stream>>>(ws + WS_SSTATS, ws + WS_OSTATS,
                                   ws + WS_SPART,  ws + WS_OPART,
                                   ws + WS_INTERB, ws + WS_PSUMB, ws + WS_PCNTB);

  // 4) finalize
  finalize<<<1, 1, 0, stream>>>(ws + WS_INTERB, ws + WS_PSUMB, ws + WS_PCNTB, out);
}